// FastGenerator_45363444580869
// MI455X (gfx1250) — compile-verified
//
#include <hip/hip_runtime.h>
#include <math.h>

// ---------------------------------------------------------------------------
// CDNA5 (gfx1250) wave32 WMMA f32 16x16x4 helpers
// ---------------------------------------------------------------------------
typedef __attribute__((ext_vector_type(2))) float v2f;
typedef __attribute__((ext_vector_type(8))) float v8f;

__device__ __forceinline__ v8f wmma4(v2f a, v2f b, v8f c) {
  // D = A(16x4) * B(4x16) + C(16x16), full f32
  return __builtin_amdgcn_wmma_f32_16x16x4_f32(false, a, false, b, (short)0, c,
                                               false, false);
}

// Each 128-thread block = 4 waves; wave w handles N-tile (blockIdx.x*4 + w).
#define WMMA_PROLOG                                                            \
  const int lane = threadIdx.x & 31;                                           \
  const int wave = threadIdx.x >> 5;                                           \
  const int r    = lane & 15;                                                  \
  const int half = lane >> 4;                                                  \
  const int m0   = blockIdx.y * 16;                                            \
  const int n0   = (blockIdx.x * 4 + wave) * 16;

// ---------------------------------------------------------------------------
// FC: out[b,o] = sum_k z[b,k]*fcw[o,k] + fcb[o].  M=Bn, N=O, K=Kd
// ---------------------------------------------------------------------------
__global__ void fc_wmma(const float* __restrict__ z, const float* __restrict__ fcw,
                        const float* __restrict__ fcb, float* __restrict__ out,
                        int N, int K) {
  WMMA_PROLOG;
  const float* arow = z + (size_t)(m0 + r) * K;
  const float* bcol = fcw + (size_t)(n0 + r) * K;
  v8f acc = {};
  for (int k = 0; k < K; k += 4) {
    v2f a, b;
    a.x = arow[k + 2 * half];     a.y = arow[k + 2 * half + 1];
    b.x = bcol[k + 2 * half];     b.y = bcol[k + 2 * half + 1];
    acc = wmma4(a, b, acc);
  }
  const int col = n0 + r;
  const float bv = fcb[col];
  for (int t = 0; t < 8; ++t)
    out[(size_t)(m0 + t + 8 * half) * N + col] = acc[t] + bv;
}

// ---------------------------------------------------------------------------
// ConvTranspose2d k4 s2 p1 as implicit GEMM.
// out[b,co,oy,ox] = sum_{ky,kx,ci} x[b,ci,iy,ix] * w[ci,co,ky,kx],
//   valid when (oy+1-ky) and (ox+1-kx) are even & in range; iy=(oy+1-ky)/2.
// M = Bn*OH*OW positions, N = Cout, K = Cin per (ky,kx) tap.
// ---------------------------------------------------------------------------
__global__ void convt_wmma(const float* __restrict__ x, const float* __restrict__ w,
                           float* __restrict__ out, int Cin, int IH, int IW,
                           int Cout) {
  WMMA_PROLOG;
  const int OH = 2 * IH, OW = 2 * IW;
  const int p = m0 + r;
  const int b = p / (OH * OW);
  const int rem = p % (OH * OW);
  const int oy = rem / OW, ox = rem % OW;
  v8f acc = {};
  const size_t aStep = (size_t)IH * IW;
  const size_t bStep = (size_t)Cout * 16;
  for (int ky = 0; ky < 4; ++ky) {
    const int ty = oy + 1 - ky;
    const bool vy = (ty >= 0) && ((ty & 1) == 0) && ((ty >> 1) < IH);
    const int iy = vy ? (ty >> 1) : 0;
    for (int kx = 0; kx < 4; ++kx) {
      const int tx = ox + 1 - kx;
      const bool vx = (tx >= 0) && ((tx & 1) == 0) && ((tx >> 1) < IW);
      const int ix = vx ? (tx >> 1) : 0;
      const bool valid = vy && vx;
      const float* ap = x + (size_t)b * Cin * aStep + (size_t)iy * IW + ix;
      const float* bp = w + (size_t)(n0 + r) * 16 + ky * 4 + kx;
      for (int k = 0; k < Cin; k += 4) {
        const int ci = k + 2 * half;
        v2f a, bb;
        a.x = valid ? ap[(size_t)ci * aStep] : 0.f;
        a.y = valid ? ap[(size_t)(ci + 1) * aStep] : 0.f;
        bb.x = bp[(size_t)ci * bStep];
        bb.y = bp[(size_t)(ci + 1) * bStep];
        acc = wmma4(a, bb, acc);
      }
    }
  }
  const int co = n0 + r;
  for (int t = 0; t < 8; ++t) {
    const int row = m0 + t + 8 * half;
    const int b2 = row / (OH * OW);
    const int r2 = row % (OH * OW);
    out[((size_t)b2 * Cout + co) * OH * OW + r2] = acc[t];
  }
}

// ---------------------------------------------------------------------------
// 1x1 conv: out[b,co,p] = bias[co] + sum_ci x[b,ci,p]*w[co,ci]
// M = Bn*HW, N = Cout, K = Cin
// ---------------------------------------------------------------------------
__global__ void conv1x1_wmma(const float* __restrict__ x, const float* __restrict__ w,
                             const float* __restrict__ bias, float* __restrict__ out,
                             int Cin, int HW, int Cout) {
  WMMA_PROLOG;
  const int m = m0 + r;
  const int b = m / HW, p = m % HW;
  const float* ap = x + (size_t)b * Cin * HW + p;
  const float* bcol = w + (size_t)(n0 + r) * Cin;
  v8f acc = {};
  for (int k = 0; k < Cin; k += 4) {
    const int ci = k + 2 * half;
    v2f a, bb;
    a.x = ap[(size_t)ci * HW];  a.y = ap[(size_t)(ci + 1) * HW];
    bb.x = bcol[ci];            bb.y = bcol[ci + 1];
    acc = wmma4(a, bb, acc);
  }
  const int co = n0 + r;
  const float bv = bias[co];
  for (int t = 0; t < 8; ++t) {
    const int row = m0 + t + 8 * half;
    const int b2 = row / HW, p2 = row % HW;
    out[((size_t)b2 * Cout + co) * HW + p2] = acc[t] + bv;
  }
}

// ---------------------------------------------------------------------------
// corr[b,c,d] = sum_n skip[b,c,n]*x[b,d,n]   (raw, normalization folded later)
// grid.z = batch; M = C (rows c), N = C (cols d), K = HW
// ---------------------------------------------------------------------------
__global__ void corr_wmma(const float* __restrict__ s, const float* __restrict__ x,
                          float* __restrict__ corr, int C, int HW) {
  WMMA_PROLOG;
  const int bz = blockIdx.z;
  const float* arow = s + ((size_t)bz * C + (m0 + r)) * HW;
  const float* bcol = x + ((size_t)bz * C + (n0 + r)) * HW;
  v8f acc = {};
  for (int k = 0; k < HW; k += 4) {
    v2f a, bb;
    a.x = arow[k + 2 * half];   a.y = arow[k + 2 * half + 1];
    bb.x = bcol[k + 2 * half];  bb.y = bcol[k + 2 * half + 1];
    acc = wmma4(a, bb, acc);
  }
  float* ob = corr + (size_t)bz * C * C;
  const int d = n0 + r;
  for (int t = 0; t < 8; ++t)
    ob[(size_t)(m0 + t + 8 * half) * C + d] = acc[t];
}

// ---------------------------------------------------------------------------
// res-mix: out[b,c,n] = x[b,c,n] + 0.3 * sum_d P[b,c,d]*x[b,d,n]
// (P already has invnorm(x) folded into columns). M=C, N=HW, K=C, grid.z=b
// ---------------------------------------------------------------------------
__global__ void resmix_wmma(const float* __restrict__ P, const float* __restrict__ x,
                            float* __restrict__ out, int C, int HW) {
  WMMA_PROLOG;
  const int bz = blockIdx.z;
  const float* arow = P + ((size_t)bz * C + (m0 + r)) * C;
  const float* xb = x + (size_t)bz * C * HW;
  float* ob = out + (size_t)bz * C * HW;
  const float* bcol = xb + (n0 + r);
  v8f acc = {};
  for (int k = 0; k < C; k += 4) {
    const int kk = k + 2 * half;
    v2f a, bb;
    a.x = arow[kk];                 a.y = arow[kk + 1];
    bb.x = bcol[(size_t)kk * HW];   bb.y = bcol[(size_t)(kk + 1) * HW];
    acc = wmma4(a, bb, acc);
  }
  const int n = n0 + r;
  for (int t = 0; t < 8; ++t) {
    const size_t idx = (size_t)(m0 + t + 8 * half) * HW + n;
    ob[idx] = xb[idx] + 0.3f * acc[t];
  }
}

// ---------------------------------------------------------------------------
// attention output: x[b,c,i] += gamma * sum_j v[b,c,j] * W[j,i]  (one batch)
// M=C(64), N=HW(1024), K=HW(1024)
// ---------------------------------------------------------------------------
__global__ void attn_out_wmma(const float* __restrict__ v, const float* __restrict__ Wat,
                              float* __restrict__ x, const float* __restrict__ gamma,
                              int b, int C, int HW) {
  WMMA_PROLOG;
  const float* arow = v + ((size_t)b * C + (m0 + r)) * HW;
  const float* bcol = Wat + (n0 + r);
  float* xb = x + (size_t)b * C * HW;
  v8f acc = {};
  for (int k = 0; k < HW; k += 4) {
    const int kk = k + 2 * half;
    v2f a, bb;
    a.x = arow[kk];                 a.y = arow[kk + 1];
    bb.x = bcol[(size_t)kk * HW];   bb.y = bcol[(size_t)(kk + 1) * HW];
    acc = wmma4(a, bb, acc);
  }
  const float g = gamma[0];
  const int n = n0 + r;
  for (int t = 0; t < 8; ++t) {
    const size_t idx = (size_t)(m0 + t + 8 * half) * HW + n;
    xb[idx] = xb[idx] + g * acc[t];
  }
}

// ---------------------------------------------------------------------------
// scalar kernels
// ---------------------------------------------------------------------------
__global__ void bn_stats(const float* __restrict__ x, float* __restrict__ mean,
                         float* __restrict__ var, int Bn, int C, int HW) {
  const int c = blockIdx.x;
  __shared__ float s1[256], s2[256];
  float a = 0.f, b2 = 0.f;
  const int n = Bn * HW;
  for (int i = threadIdx.x; i < n; i += 256) {
    const int b = i / HW, p = i % HW;
    const float v = x[((size_t)b * C + c) * HW + p];
    a += v; b2 += v * v;
  }
  s1[threadIdx.x] = a; s2[threadIdx.x] = b2; __syncthreads();
  for (int o = 128; o > 0; o >>= 1) {
    if (threadIdx.x < o) { s1[threadIdx.x] += s1[threadIdx.x + o];
                           s2[threadIdx.x] += s2[threadIdx.x + o]; }
    __syncthreads();
  }
  if (threadIdx.x == 0) {
    const float m = s1[0] / n;
    mean[c] = m;
    var[c]  = fmaxf(s2[0] / n - m * m, 0.f);
  }
}

__global__ void bn_lrelu(float* __restrict__ x, const float* __restrict__ mean,
                         const float* __restrict__ var, const float* __restrict__ g,
                         const float* __restrict__ bta, int C, int HW, int total) {
  const int idx = blockIdx.x * 256 + threadIdx.x;
  if (idx >= total) return;
  const int c = (idx / HW) % C;
  const float y = g[c] * (x[idx] - mean[c]) * rsqrtf(var[c] + 1e-5f) + bta[c];
  x[idx] = (y > 0.f) ? y : 0.2f * y;
}

__global__ void qk_conv(const float* __restrict__ x, const float* __restrict__ w,
                        const float* __restrict__ bias, float* __restrict__ out,
                        int Cin, int HW, int Cout, int total) {
  const int idx = blockIdx.x * 256 + threadIdx.x;
  if (idx >= total) return;
  const int p = idx % HW;
  const int t = idx / HW;
  const int co = t % Cout, b = t / Cout;
  const float* xp = x + (size_t)b * Cin * HW + p;
  const float* wp = w + (size_t)co * Cin;
  float s = bias[co];
  for (int ci = 0; ci < Cin; ++ci) s += xp[(size_t)ci * HW] * wp[ci];
  out[idx] = s;
}

// column-softmax attention weights for one batch:
// W[j,i] = softmax_i( sum_c q[b,c,i]*k[b,c,j] ).  HW=1024, 256 thr, 4 i/thr.
__global__ void attn_w_kernel(const float* __restrict__ q, const float* __restrict__ k,
                              float* __restrict__ Wat, int b, int HW) {
  const int j = blockIdx.x;
  const float* qb = q + (size_t)b * 8 * HW;
  const float* kb = k + (size_t)b * 8 * HW;
  float kv[8];
  for (int c = 0; c < 8; ++c) kv[c] = kb[(size_t)c * HW + j];
  __shared__ float red[256];
  float sv[4]; float mx = -1e30f;
  for (int t = 0; t < 4; ++t) {
    const int i = threadIdx.x + t * 256;
    float s = 0.f;
    for (int c = 0; c < 8; ++c) s += qb[(size_t)c * HW + i] * kv[c];
    sv[t] = s; mx = fmaxf(mx, s);
  }
  red[threadIdx.x] = mx; __syncthreads();
  for (int o = 128; o > 0; o >>= 1) {
    if (threadIdx.x < o) red[threadIdx.x] = fmaxf(red[threadIdx.x], red[threadIdx.x + o]);
    __syncthreads();
  }
  mx = red[0]; __syncthreads();
  float sum = 0.f;
  for (int t = 0; t < 4; ++t) { sv[t] = expf(sv[t] - mx); sum += sv[t]; }
  red[threadIdx.x] = sum; __syncthreads();
  for (int o = 128; o > 0; o >>= 1) {
    if (threadIdx.x < o) red[threadIdx.x] += red[threadIdx.x + o];
    __syncthreads();
  }
  const float inv = 1.f / red[0];
  for (int t = 0; t < 4; ++t)
    Wat[(size_t)j * HW + threadIdx.x + t * 256] = sv[t] * inv;
}

// bilinear x2 upsample, half-pixel centers, edge clamp
__global__ void resize2x(const float* __restrict__ in, float* __restrict__ out,
                         int H, int W, int total) {
  const int idx = blockIdx.x * 256 + threadIdx.x;
  if (idx >= total) return;
  const int OW = 2 * W, OH = 2 * H;
  const int ox = idx % OW;
  int t = idx / OW;
  const int oy = t % OH;
  const int bc = t / OH;
  const float sy = 0.5f * (oy + 0.5f) - 0.5f;
  const float sx = 0.5f * (ox + 0.5f) - 0.5f;
  const int y0 = (int)floorf(sy), x0 = (int)floorf(sx);
  const float fy = sy - y0, fx = sx - x0;
  const int y0c = min(max(y0, 0), H - 1), y1c = min(max(y0 + 1, 0), H - 1);
  const int x0c = min(max(x0, 0), W - 1), x1c = min(max(x0 + 1, 0), W - 1);
  const float* p = in + (size_t)bc * H * W;
  const float v00 = p[y0c * W + x0c], v01 = p[y0c * W + x1c];
  const float v10 = p[y1c * W + x0c], v11 = p[y1c * W + x1c];
  out[idx] = v00 * (1.f - fy) * (1.f - fx) + v01 * (1.f - fy) * fx +
             v10 * fy * (1.f - fx) + v11 * fy * fx;
}

__global__ void row_invnorm(const float* __restrict__ x, float* __restrict__ invn,
                            int HW) {
  const int row = blockIdx.x;
  __shared__ float red[256];
  float a = 0.f;
  const float* p = x + (size_t)row * HW;
  for (int i = threadIdx.x; i < HW; i += 256) { const float v = p[i]; a += v * v; }
  red[threadIdx.x] = a; __syncthreads();
  for (int o = 128; o > 0; o >>= 1) {
    if (threadIdx.x < o) red[threadIdx.x] += red[threadIdx.x + o];
    __syncthreads();
  }
  if (threadIdx.x == 0) invn[row] = 1.f / fmaxf(sqrtf(red[0]), 1e-12f);
}

// corr row: scale by invnorms + 1/sqrt(hw), softmax over d, fold invx[d] back
__global__ void corr_softmax(float* __restrict__ corr, const float* __restrict__ invs,
                             const float* __restrict__ invx, int C, float rscale) {
  const int c = blockIdx.x, b = blockIdx.y;
  float* row = corr + ((size_t)b * C + c) * C;
  const float* isb = invs + (size_t)b * C;
  const float* ixb = invx + (size_t)b * C;
  const int d = threadIdx.x;
  __shared__ float red[256];
  float v = 0.f, key = -1e30f;
  if (d < C) { v = row[d] * isb[c] * ixb[d] * rscale; key = v; }
  red[d] = key; __syncthreads();
  for (int o = 128; o > 0; o >>= 1) {
    if (d < o) red[d] = fmaxf(red[d], red[d + o]);
    __syncthreads();
  }
  const float mx = red[0]; __syncthreads();
  const float e = (d < C) ? expf(v - mx) : 0.f;
  red[d] = e; __syncthreads();
  for (int o = 128; o > 0; o >>= 1) {
    if (d < o) red[d] += red[d + o];
    __syncthreads();
  }
  if (d < C) row[d] = (e / red[0]) * ixb[d];
}

// final convT (Cout=3) + tanh, scalar
__global__ void final_convt_tanh(const float* __restrict__ x, const float* __restrict__ w,
                                 float* __restrict__ out, int Cin, int IH, int IW,
                                 int Cout, int total) {
  const int idx = blockIdx.x * 256 + threadIdx.x;
  if (idx >= total) return;
  const int OH = 2 * IH, OW = 2 * IW;
  const int ox = idx % OW;
  int t = idx / OW;
  const int oy = t % OH; t /= OH;
  const int co = t % Cout;
  const int b = t / Cout;
  float s = 0.f;
  for (int ky = 0; ky < 4; ++ky) {
    const int ty = oy + 1 - ky;
    if (ty < 0 || (ty & 1)) continue;
    const int iy = ty >> 1; if (iy >= IH) continue;
    for (int kx = 0; kx < 4; ++kx) {
      const int tx = ox + 1 - kx;
      if (tx < 0 || (tx & 1)) continue;
      const int ix = tx >> 1; if (ix >= IW) continue;
      const float* xp = x + (size_t)b * Cin * IH * IW + iy * IW + ix;
      const float* wp = w + (size_t)co * 16 + ky * 4 + kx;
      for (int ci = 0; ci < Cin; ++ci)
        s += xp[(size_t)ci * IH * IW] * wp[(size_t)ci * Cout * 16];
    }
  }
  out[idx] = tanhf(s);
}

// ---------------------------------------------------------------------------
// host orchestration
// ---------------------------------------------------------------------------
extern "C" void kernel_launch(void* const* d_in, const int* in_sizes, int n_in,
                              void* d_out, int out_size, void* d_ws, size_t ws_size,
                              hipStream_t stream) {
  const float* z    = (const float*)d_in[0];
  const float* fc_w = (const float*)d_in[1];
  const float* fc_b = (const float*)d_in[2];
  const float* wS[3]  = {(const float*)d_in[3], (const float*)d_in[8],  (const float*)d_in[13]};
  const float* gS[3]  = {(const float*)d_in[4], (const float*)d_in[9],  (const float*)d_in[14]};
  const float* bS[3]  = {(const float*)d_in[5], (const float*)d_in[10], (const float*)d_in[15]};
  const float* swS[3] = {(const float*)d_in[6], (const float*)d_in[11], (const float*)d_in[16]};
  const float* sbS[3] = {(const float*)d_in[7], (const float*)d_in[12], (const float*)d_in[17]};
  const float* wq = (const float*)d_in[18];
  const float* bq = (const float*)d_in[19];
  const float* wk = (const float*)d_in[20];
  const float* bk = (const float*)d_in[21];
  const float* wv = (const float*)d_in[22];
  const float* bv = (const float*)d_in[23];
  const float* gamma = (const float*)d_in[24];
  const float* w3 = (const float*)d_in[25];

  float* Warena = (float*)d_ws;
  size_t off = 0;
  auto alloc = [&](size_t n) { float* p = Warena + off; off += n; return p; };
  float* x0     = alloc(64 * 8192);        // FC output = (64,512,4,4)
  float* convb  = alloc(64 * 64 * 1024);   // conv/BN/attn buffer (max 4.19M)
  float* xA     = alloc(64 * 64 * 1024);
  float* xB     = alloc(64 * 64 * 1024);
  float* skpre  = alloc(64 * 64 * 256);    // skip pre-resize (max 1.05M)
  float* skpost = alloc(64 * 64 * 1024);
  float* corr   = alloc(64 * 256 * 256);   // also aliases attention V (disjoint lifetime)
  float* vbuf   = corr;
  float* qbuf   = alloc(64 * 8 * 1024);
  float* kbuf   = alloc(64 * 8 * 1024);
  float* Wat    = alloc(1024 * 1024);      // per-batch attention weights
  float* meanb  = alloc(256);
  float* varb   = alloc(256);
  float* invs   = alloc(64 * 256);
  float* invx   = alloc(64 * 256);

  const dim3 blk(128);

  // FC: M=64, N=8192, K=100
  fc_wmma<<<dim3(8192 / 64, 64 / 16), blk, 0, stream>>>(z, fc_w, fc_b, x0, 8192, 100);

  const int CinS[3]  = {512, 256, 128};
  const int CoutS[3] = {256, 128, 64};
  const int IHS[3]   = {4, 8, 16};
  float* xmix[3] = {xA, xB, xA};
  const float* xin = x0;

  for (int s = 0; s < 3; ++s) {
    const int Cin = CinS[s], Cout = CoutS[s], IH = IHS[s], IW = IH;
    const int OH = 2 * IH, OW = 2 * IW;
    const int HWi = IH * IW, HWo = OH * OW;
    const int Mi = 64 * HWi, Mo = 64 * HWo;
    const int tot = 64 * Cout * HWo;

    // skip = 1x1 conv of feat at input resolution
    conv1x1_wmma<<<dim3(Cout / 64, Mi / 16), blk, 0, stream>>>(
        xin, swS[s], sbS[s], skpre, Cin, HWi, Cout);

    // transposed conv (implicit GEMM)
    convt_wmma<<<dim3(Cout / 64, Mo / 16), blk, 0, stream>>>(
        xin, wS[s], convb, Cin, IH, IW, Cout);

    // BatchNorm (batch stats) + LeakyReLU(0.2), in place
    bn_stats<<<Cout, 256, 0, stream>>>(convb, meanb, varb, 64, Cout, HWo);
    bn_lrelu<<<(tot + 255) / 256, 256, 0, stream>>>(convb, meanb, varb, gS[s], bS[s],
                                                    Cout, HWo, tot);

    if (s == 2) {  // self-attention at 32x32, C=64
      const int HW = 1024, totqk = 64 * 8 * HW;
      qk_conv<<<(totqk + 255) / 256, 256, 0, stream>>>(convb, wq, bq, qbuf, 64, HW, 8, totqk);
      qk_conv<<<(totqk + 255) / 256, 256, 0, stream>>>(convb, wk, bk, kbuf, 64, HW, 8, totqk);
      conv1x1_wmma<<<dim3(1, (64 * HW) / 16), blk, 0, stream>>>(convb, wv, bv, vbuf,
                                                                64, HW, 64);
      for (int b = 0; b < 64; ++b) {  // stream-serialized per-batch attention
        attn_w_kernel<<<HW, 256, 0, stream>>>(qbuf, kbuf, Wat, b, HW);
        attn_out_wmma<<<dim3(HW / 64, 64 / 16), blk, 0, stream>>>(vbuf, Wat, convb,
                                                                  gamma, b, 64, HW);
      }
    }

    // bilinear upsample skip to output resolution
    resize2x<<<(tot + 255) / 256, 256, 0, stream>>>(skpre, skpost, IH, IW, tot);

    // skip-mix: norms -> raw corr GEMM -> scaled softmax -> res GEMM (+0.3 mix)
    row_invnorm<<<64 * Cout, 256, 0, stream>>>(skpost, invs, HWo);
    row_invnorm<<<64 * Cout, 256, 0, stream>>>(convb, invx, HWo);
    corr_wmma<<<dim3(Cout / 64, Cout / 16, 64), blk, 0, stream>>>(skpost, convb, corr,
                                                                  Cout, HWo);
    corr_softmax<<<dim3(Cout, 64), 256, 0, stream>>>(corr, invs, invx, Cout,
                                                     1.0f / sqrtf((float)HWo));
    resmix_wmma<<<dim3(HWo / 64, Cout / 16, 64), blk, 0, stream>>>(corr, convb,
                                                                   xmix[s], Cout, HWo);
    xin = xmix[s];
  }

  // final transposed conv (Cout=3) + tanh
  const int totf = 64 * 3 * 64 * 64;
  final_convt_tanh<<<(totf + 255) / 256, 256, 0, stream>>>(xin, w3, (float*)d_out,
                                                           64, 32, 32, 3, totf);
  (void)in_sizes; (void)n_in; (void)out_size; (void)ws_size;
}